// Proposal_28355374088476
// MI455X (gfx1250) — compile-verified
//
#include <hip/hip_runtime.h>
#include <hip/hip_bf16.h>
#include <stdint.h>

#define A_NUM   131072
#define BATCH   4
#define PRE     3000
#define MAXOUT  1000
#define NMS_THR 0.7f
#define NWORDS  94          // ceil(3000/32)
#define CAND_CAP 4096

// ---- workspace byte offsets ----
#define OFF_HIST1   0u          // 4*65536*4  = 1048576
#define OFF_HIST2   1048576u    // 4*65536*4  = 1048576
#define OFF_MISC    2097152u    // 64 u32
#define OFF_CAND    2097408u    // 4*4096*8   = 131072
#define OFF_TOPIDX  2228480u    // 4*3000*4   = 48000
#define OFF_BOXES   2276480u    // 4*3000*6*4 = 288000
#define OFF_SCORES  2564480u    // 4*3000*4   = 48000
#define OFF_KEEP    2612480u    // 4*1000*4   = 16000
#define OFF_MASK    2628608u    // 4*3000*94*4 = 4512000  (total ~7.14 MB)

// misc u32 slots
#define MI_THRB 0   // hi16 threshold bin, per batch
#define MI_CUMA 4   // count of keys with hi16 > THRB
#define MI_THRT 8   // final 32-bit key threshold
#define MI_CNT  12  // candidate counters
#define MI_NK   16  // number kept (<=1000)

__device__ __forceinline__ uint32_t fkey(float f) {
    uint32_t u = __float_as_uint(f);
    return (u & 0x80000000u) ? ~u : (u | 0x80000000u);   // monotonic: bigger float -> bigger key
}

// ---------------- zero the regions that must start at 0 ----------------
__global__ void clear_kernel(uint32_t* ws32, int n) {
    for (int i = blockIdx.x * blockDim.x + threadIdx.x; i < n; i += gridDim.x * blockDim.x)
        ws32[i] = 0u;
}

// ---------------- pass 1: histogram over hi16 of key ----------------
__global__ void hist1_kernel(const float* __restrict__ scores, uint32_t* __restrict__ hist1) {
    int g = blockIdx.x * blockDim.x + threadIdx.x;
    if (g >= BATCH * A_NUM) return;
    int b = g >> 17;                       // /131072
    uint32_t key = fkey(scores[g]);
    atomicAdd(&hist1[b * 65536 + (key >> 16)], 1u);
}

// Parallel threshold finder over a 65536-bin histogram:
// finds bin B with countAbove(B) < target <= countAbove(B) + h[B];
// returns B via *binOut and countAbove(B) via *aboveOut. Block of 256 threads.
__device__ __forceinline__ void find_threshold_256(const uint32_t* __restrict__ h,
                                                   uint32_t target,
                                                   uint32_t* __restrict__ binOut,
                                                   uint32_t* __restrict__ aboveOut) {
    __shared__ uint32_t buf[256];
    __shared__ uint32_t sChunk, sBase;
    int tid = threadIdx.x;

    // ---- level 1: per-chunk sums (chunk = 256 bins) ----
    uint32_t s = 0;
    for (int j = 0; j < 256; j++) s += h[tid * 256 + j];
    buf[tid] = s;
    __syncthreads();
    // suffix sum: buf[c] = sum of chunk sums c..255
    for (int off = 1; off < 256; off <<= 1) {
        uint32_t v = buf[tid] + ((tid + off < 256) ? buf[tid + off] : 0u);
        __syncthreads();
        buf[tid] = v;
        __syncthreads();
    }
    // unique c: suffix[c] >= target and suffix[c+1] < target
    {
        uint32_t sfx  = buf[tid];
        uint32_t sfx1 = (tid + 1 < 256) ? buf[tid + 1] : 0u;
        if (sfx >= target && sfx1 < target) { sChunk = (uint32_t)tid; sBase = sfx1; }
    }
    __syncthreads();
    uint32_t c = sChunk, base1 = sBase;
    __syncthreads();

    // ---- level 2: bins within chunk c ----
    buf[tid] = h[c * 256 + tid];
    __syncthreads();
    for (int off = 1; off < 256; off <<= 1) {
        uint32_t v = buf[tid] + ((tid + off < 256) ? buf[tid + off] : 0u);
        __syncthreads();
        buf[tid] = v;
        __syncthreads();
    }
    {
        uint32_t sfx  = base1 + buf[tid];
        uint32_t sfx1 = base1 + ((tid + 1 < 256) ? buf[tid + 1] : 0u);
        if (sfx >= target && sfx1 < target) {
            *binOut   = c * 256 + (uint32_t)tid;
            *aboveOut = sfx1;
        }
    }
}

// find hi16 bin B: count(hi>B) < PRE <= count(hi>=B)
__global__ void scan1_kernel(const uint32_t* __restrict__ hist, uint32_t* __restrict__ misc) {
    int b = blockIdx.x;
    find_threshold_256(hist + b * 65536, PRE, &misc[MI_THRB + b], &misc[MI_CUMA + b]);
}

// ---------------- pass 2: histogram over lo16, only for hi16 == B ----------------
__global__ void hist2_kernel(const float* __restrict__ scores, const uint32_t* __restrict__ misc,
                             uint32_t* __restrict__ hist2) {
    int g = blockIdx.x * blockDim.x + threadIdx.x;
    if (g >= BATCH * A_NUM) return;
    int b = g >> 17;
    uint32_t key = fkey(scores[g]);
    if ((key >> 16) == misc[MI_THRB + b])
        atomicAdd(&hist2[b * 65536 + (key & 0xFFFFu)], 1u);
}

__global__ void scan2_kernel(const uint32_t* __restrict__ hist, uint32_t* __restrict__ misc) {
    int b = blockIdx.x;
    __shared__ uint32_t lbin, labove;
    uint32_t target = PRE - misc[MI_CUMA + b];   // >= 1
    find_threshold_256(hist + b * 65536, target, &lbin, &labove);
    __syncthreads();
    if (threadIdx.x == 0)
        misc[MI_THRT + b] = (misc[MI_THRB + b] << 16) | lbin;
}

// ---------------- collect candidates: key >= threshold ----------------
__global__ void collect_kernel(const float* __restrict__ scores, uint32_t* __restrict__ misc,
                               unsigned long long* __restrict__ cand) {
    int g = blockIdx.x * blockDim.x + threadIdx.x;
    if (g >= BATCH * A_NUM) return;
    int b = g >> 17;
    int idx = g & (A_NUM - 1);
    uint32_t key = fkey(scores[g]);
    if (key >= misc[MI_THRT + b]) {
        uint32_t pos = atomicAdd(&misc[MI_CNT + b], 1u);
        if (pos < CAND_CAP) {
            // composite: key desc primary, index asc secondary (under descending sort)
            unsigned long long c = ((unsigned long long)key << 32) |
                                   (unsigned long long)(0xFFFFFFFFu - (uint32_t)idx);
            cand[b * CAND_CAP + pos] = c;
        }
    }
}

// ---------------- bitonic sort 4096 (descending), emit top-3000 indices ----------------
__global__ __launch_bounds__(1024) void sort_kernel(const unsigned long long* __restrict__ cand,
                                                    const uint32_t* __restrict__ misc,
                                                    uint32_t* __restrict__ topidx) {
    int b = blockIdx.x, tid = threadIdx.x;
    __shared__ unsigned long long sm[CAND_CAP];
    uint32_t cnt = misc[MI_CNT + b]; if (cnt > CAND_CAP) cnt = CAND_CAP;
    for (int i = tid; i < CAND_CAP; i += 1024)
        sm[i] = (i < (int)cnt) ? cand[b * CAND_CAP + i] : 0ull;
    __syncthreads();
    for (uint32_t k = 2; k <= CAND_CAP; k <<= 1) {
        for (uint32_t j = k >> 1; j > 0; j >>= 1) {
            for (uint32_t i = tid; i < CAND_CAP; i += 1024) {
                uint32_t l = i ^ j;
                if (l > i) {
                    unsigned long long a = sm[i], c = sm[l];
                    bool up = ((i & k) == 0);
                    if (up ? (a < c) : (a > c)) { sm[i] = c; sm[l] = a; }
                }
            }
            __syncthreads();
        }
    }
    for (int i = tid; i < PRE; i += 1024)
        topidx[b * PRE + i] = 0xFFFFFFFFu - (uint32_t)(sm[i] & 0xFFFFFFFFull);
}

// ---------------- box regression + score gather for the top-3000 ----------------
__global__ void regress_kernel(const float* __restrict__ anchors, const float* __restrict__ scores,
                               const float* __restrict__ deltas, const uint32_t* __restrict__ topidx,
                               float* __restrict__ boxes_s, float* __restrict__ scores_s) {
    int t = blockIdx.x * blockDim.x + threadIdx.x;
    if (t >= BATCH * PRE) return;
    int b = t / PRE, i = t - b * PRE;
    uint32_t a = topidx[b * PRE + i];
    const float* A = anchors + (size_t)a * 6;
    const float* D = deltas + ((size_t)b * A_NUM + a) * 6;
    float* Bo = boxes_s + (size_t)(b * PRE + i) * 6;
#pragma unroll
    for (int d = 0; d < 3; d++) {
        float l = A[d], h = A[d + 3];
        float dim = h - l;
        float ctr = l + 0.5f * dim + D[d] * dim;
        float nd  = dim * expf(D[d + 3]);
        Bo[d]     = ctr - 0.5f * nd;
        Bo[d + 3] = ctr + 0.5f * nd;
    }
    scores_s[b * PRE + i] = scores[(size_t)b * A_NUM + a];
}

// ---------------- IoU bitmask kernel: 32 rows x 32 cols per one-wave block ----------------
// Column boxes (768 B) are staged into LDS with CDNA5 async global->LDS copies.
__global__ void mask_kernel(const float* __restrict__ boxes_s, uint32_t* __restrict__ mask) {
    int bx = blockIdx.x, by = blockIdx.y, b = blockIdx.z;
    if (bx < by) return;                       // only need cols >= row block
    int tid = threadIdx.x;                     // 32 threads = 1 wave
    __shared__ __align__(16) float scol[32 * 6];

    const float* colbase = boxes_s + ((size_t)b * PRE + (size_t)bx * 32) * 6;
    uint32_t laddr = (uint32_t)(uintptr_t)(&scol[0]) + (uint32_t)(tid * 24);
    uint64_t gaddr = (uint64_t)(uintptr_t)colbase + (uint64_t)(tid * 24);
    // 3 x b64 per lane = 24B/lane = 768B total, tracked by ASYNCcnt
    asm volatile(
        "global_load_async_to_lds_b64 %0, %1, off\n"
        "global_load_async_to_lds_b64 %0, %1, off offset:8\n"
        "global_load_async_to_lds_b64 %0, %1, off offset:16\n"
        :: "v"(laddr), "v"(gaddr) : "memory");
    asm volatile("s_wait_asynccnt 0" ::: "memory");
    __syncthreads();

    int i = by * 32 + tid;
    if (i >= PRE) return;
    const float* R = boxes_s + ((size_t)b * PRE + i) * 6;
    float rl0 = R[0], rl1 = R[1], rl2 = R[2], rh0 = R[3], rh1 = R[4], rh2 = R[5];
    float rv = fmaxf(rh0 - rl0, 0.f) * fmaxf(rh1 - rl1, 0.f) * fmaxf(rh2 - rl2, 0.f);
    uint32_t bits = 0u;
    int ncol = PRE - bx * 32; if (ncol > 32) ncol = 32;
    for (int j = 0; j < ncol; j++) {
        const float* C = &scol[j * 6];
        float d0 = fmaxf(fminf(rh0, C[3]) - fmaxf(rl0, C[0]), 0.f);
        float d1 = fmaxf(fminf(rh1, C[4]) - fmaxf(rl1, C[1]), 0.f);
        float d2 = fmaxf(fminf(rh2, C[5]) - fmaxf(rl2, C[2]), 0.f);
        float iv = d0 * d1 * d2;
        float cv = fmaxf(C[3] - C[0], 0.f) * fmaxf(C[4] - C[1], 0.f) * fmaxf(C[5] - C[2], 0.f);
        float iou = iv / fmaxf(rv + cv - iv, 1e-8f);
        if (iou > NMS_THR) bits |= (1u << j);
    }
    mask[(size_t)(b * PRE + i) * NWORDS + bx] = bits;
}

// ---------------- sequential NMS reduce: one wave per batch ----------------
__global__ void nms_reduce_kernel(const uint32_t* __restrict__ mask,
                                  uint32_t* __restrict__ keep_list,
                                  uint32_t* __restrict__ misc) {
    int b = blockIdx.x, tid = threadIdx.x;     // 32 threads
    uint32_t r0 = 0, r1 = 0, r2 = 0;           // 94 suppression words across the wave
    uint32_t nk = 0;
    for (int i = 0; i < PRE; i++) {
        const uint32_t* row = mask + (size_t)(b * PRE + i) * NWORDS;
        if (i + 1 < PRE)  // hide next-row latency (global_prefetch_b8)
            __builtin_prefetch(mask + (size_t)(b * PRE + i + 1) * NWORDS + tid, 0, 0);
        int word = i >> 5, bit = i & 31;
        uint32_t rw = (word < 32) ? __shfl(r0, word, 32)
                    : (word < 64) ? __shfl(r1, word - 32, 32)
                                  : __shfl(r2, word - 64, 32);
        if (((rw >> bit) & 1u) == 0u) {        // box i kept
            r0 |= row[tid];
            r1 |= row[tid + 32];               // words 32..63
            if (tid < NWORDS - 64) r2 |= row[tid + 64];  // words 64..93
            if (tid == 0 && nk < MAXOUT) keep_list[b * MAXOUT + nk] = (uint32_t)i;
            nk++;
        }
    }
    if (tid == 0) misc[MI_NK + b] = (nk < MAXOUT) ? nk : MAXOUT;
}

// ---------------- write outputs: props | scores | batch-index | valid ----------------
__global__ void output_kernel(const float* __restrict__ boxes_s, const float* __restrict__ scores_s,
                              const uint32_t* __restrict__ keep_list, const uint32_t* __restrict__ misc,
                              float* __restrict__ out) {
    int t = blockIdx.x * blockDim.x + threadIdx.x;
    if (t >= BATCH * MAXOUT) return;
    int b = t / MAXOUT, k = t - b * MAXOUT;
    float* props = out;
    float* sc  = out + BATCH * MAXOUT * 6;
    float* ind = sc + BATCH * MAXOUT;
    float* val = ind + BATCH * MAXOUT;
    uint32_t nk = misc[MI_NK + b];
    if ((uint32_t)k < nk) {
        uint32_t i = keep_list[b * MAXOUT + k];
        const float* Bx = boxes_s + (size_t)(b * PRE + i) * 6;
#pragma unroll
        for (int d = 0; d < 6; d++) props[t * 6 + d] = Bx[d];
        sc[t] = scores_s[b * PRE + i];
        val[t] = 1.0f;
    } else {
#pragma unroll
        for (int d = 0; d < 6; d++) props[t * 6 + d] = 0.0f;
        sc[t] = 0.0f;
        val[t] = 0.0f;
    }
    ind[t] = (float)b;
}

extern "C" void kernel_launch(void* const* d_in, const int* in_sizes, int n_in,
                              void* d_out, int out_size, void* d_ws, size_t ws_size,
                              hipStream_t stream) {
    const float* anchors = (const float*)d_in[0];   // (131072, 6)
    const float* scores  = (const float*)d_in[1];   // (4, 131072)
    const float* deltas  = (const float*)d_in[2];   // (4, 131072, 6)
    float* out = (float*)d_out;

    uint8_t* ws = (uint8_t*)d_ws;
    uint32_t* hist1   = (uint32_t*)(ws + OFF_HIST1);
    uint32_t* hist2   = (uint32_t*)(ws + OFF_HIST2);
    uint32_t* misc    = (uint32_t*)(ws + OFF_MISC);
    unsigned long long* cand = (unsigned long long*)(ws + OFF_CAND);
    uint32_t* topidx  = (uint32_t*)(ws + OFF_TOPIDX);
    float* boxes_s    = (float*)(ws + OFF_BOXES);
    float* scores_s   = (float*)(ws + OFF_SCORES);
    uint32_t* keepl   = (uint32_t*)(ws + OFF_KEEP);
    uint32_t* mask    = (uint32_t*)(ws + OFF_MASK);

    // re-zero histograms + counters every call (graph replays must be identical)
    clear_kernel<<<512, 256, 0, stream>>>((uint32_t*)ws, (int)(OFF_CAND / 4));

    int ng = (BATCH * A_NUM) / 256;
    hist1_kernel<<<ng, 256, 0, stream>>>(scores, hist1);
    scan1_kernel<<<BATCH, 256, 0, stream>>>(hist1, misc);
    hist2_kernel<<<ng, 256, 0, stream>>>(scores, misc, hist2);
    scan2_kernel<<<BATCH, 256, 0, stream>>>(hist2, misc);
    collect_kernel<<<ng, 256, 0, stream>>>(scores, misc, cand);
    sort_kernel<<<BATCH, 1024, 0, stream>>>(cand, misc, topidx);
    regress_kernel<<<(BATCH * PRE + 255) / 256, 256, 0, stream>>>(anchors, scores, deltas,
                                                                  topidx, boxes_s, scores_s);
    mask_kernel<<<dim3(NWORDS, NWORDS, BATCH), 32, 0, stream>>>(boxes_s, mask);
    nms_reduce_kernel<<<BATCH, 32, 0, stream>>>(mask, keepl, misc);
    output_kernel<<<(BATCH * MAXOUT + 255) / 256, 256, 0, stream>>>(boxes_s, scores_s, keepl,
                                                                    misc, out);
    (void)in_sizes; (void)n_in; (void)out_size; (void)ws_size;
}